// Adapter_30674656428557
// MI455X (gfx1250) — compile-verified
//
#include <hip/hip_runtime.h>

// ---------------------------------------------------------------------------
// GNN temporal adapter for MI455X (gfx1250), f32 end-to-end via
// V_WMMA_F32_16X16X4_F32. Memory-bound workload (~315 MB HBM, ~10.6 GFLOP):
// the key optimization is fusing time-projection+ReLU+scatter into one pass so
// the 409.6 MB time_feat tensor is never materialized; the [N,64] accumulator
// (12.8 MB) lives in the 192 MB L2 and absorbs all scatter atomics.
//
// Round-3 change: tile-base and tail predicates are forced into SGPRs with
// readfirstlane (they are wave-uniform by construction, but the compiler
// cannot prove it from threadIdx.x>>5), so full/tail selection becomes a
// scalar s_cbranch and the hot path issues straight-line atomics/stores with
// no exec-mask juggling.
// ---------------------------------------------------------------------------

#define IN_CH    256
#define ADAPTER  64
#define EDGE_DIM 32

typedef __attribute__((ext_vector_type(2))) float v2f;
typedef __attribute__((ext_vector_type(8))) float v8f;

// D = A(16x4 f32) * B(4x16 f32) + C(16x16 f32)
// A layout: lanes 0-15 -> M=lane, K=0,1 in v[0],v[1]; lanes 16-31 -> K=2,3.
// B layout (mirror): lanes 0-15 -> N=lane, K=0,1; lanes 16-31 -> K=2,3.
// C layout: VGPR r -> M=r (lanes 0-15) / M=r+8 (lanes 16-31), N = lane%16.
__device__ __forceinline__ v8f wmma_f32(v2f a, v2f b, v8f c) {
  return __builtin_amdgcn_wmma_f32_16x16x4_f32(
      /*neg_a=*/false, a, /*neg_b=*/false, b,
      /*c_mod=*/(short)0, c, /*reuse_a=*/false, /*reuse_b=*/false);
}

__device__ __forceinline__ float frelu(float v) { return v > 0.f ? v : 0.f; }

// Wave-uniform value -> SGPR (value must be uniform across the wave).
__device__ __forceinline__ int uni(int v) {
  return __builtin_amdgcn_readfirstlane(v);
}

// ---------------------------------------------------------------------------
// Zero the scatter accumulator (sums [N,64] followed by counts [N]).
// ---------------------------------------------------------------------------
__global__ void zero_kernel(float* __restrict__ p, int n) {
  int i = blockIdx.x * blockDim.x + threadIdx.x;
  if (i < n) p[i] = 0.f;
}

// ---------------------------------------------------------------------------
// Edge kernel: time_feat = relu(edge_attr @ W_time^T + b_time), scatter-add
// into sums[src] and counts[src]. One 16-edge tile per wave, K=32 in 8 WMMA
// steps, 4 N-tiles of 16 output channels. 32 v_wmma per tile.
// ---------------------------------------------------------------------------
__global__ __launch_bounds__(256) void edge_time_scatter(
    const float* __restrict__ edge_attr,  // [E, 32]
    const int*   __restrict__ edge_src,   // edge_index row 0, [E]
    const float* __restrict__ W_time,     // [64, 32]
    const float* __restrict__ b_time,     // [64]
    float*       __restrict__ sums,       // [N, 64]
    float*       __restrict__ counts,     // [N]
    int n_edges)
{
  const int lane  = threadIdx.x & 31;
  const int wave  = threadIdx.x >> 5;
  const int half  = lane >> 4;   // 0: K pair {0,1}; 1: K pair {2,3}
  const int idx16 = lane & 15;   // M (A-frag) or N (B-frag / C column)

  // Scalarized tile base: uniform within the wave, now provably so.
  const int e0 = uni((blockIdx.x * 8 + wave) * 16);
  if (e0 >= n_edges) return;                // no barriers in this kernel

  // Scalar tail predicate: the whole 16-edge tile is in bounds except for at
  // most one tail tile in the entire grid.
  const bool full = uni(e0 + 16 <= n_edges) != 0;

  // A fragments for all 8 K-steps: one edge_attr row per lane-16-group.
  const int ea = full ? (e0 + idx16) : min(e0 + idx16, n_edges - 1);
  v2f afrag[8];
  {
    const float* row = edge_attr + ea * EDGE_DIM + half * 2;
#pragma unroll
    for (int s = 0; s < 8; ++s) {
      afrag[s].x = row[s * 4 + 0];
      afrag[s].y = row[s * 4 + 1];
    }
  }

  // Source node of the 8 edges whose C-rows this lane holds (L0-cached,
  // broadcast across the 16-lane group).
  int esrc[8];
#pragma unroll
  for (int r = 0; r < 8; ++r) {
    const int e = e0 + half * 8 + r;
    esrc[r] = edge_src[full ? e : min(e, n_edges - 1)];
  }

  // Degree count: one atomic per edge, issued by lanes 0..15.
  if (full) {
    if (half == 0) atomicAdd(&counts[edge_src[e0 + idx16]], 1.0f);
  } else {
    if (half == 0 && (e0 + idx16) < n_edges)
      atomicAdd(&counts[edge_src[e0 + idx16]], 1.0f);
  }

#pragma unroll
  for (int nt = 0; nt < 4; ++nt) {          // output-channel tiles
    v8f c = {};
#pragma unroll
    for (int s = 0; s < 8; ++s) {           // K steps of 4
      const int k = s * 4 + half * 2;
      const float* wr = W_time + (nt * 16 + idx16) * EDGE_DIM + k;
      v2f b; b.x = wr[0]; b.y = wr[1];      // B[k][n] = W_time[n][k]
      c = wmma_f32(afrag[s], b, c);
    }
    const int   ch   = nt * 16 + idx16;
    const float bias = b_time[ch];
    if (full) {
      // Straight-line atomic issue; accumulator is L2-resident.
#pragma unroll
      for (int r = 0; r < 8; ++r)
        atomicAdd(&sums[esrc[r] * ADAPTER + ch], frelu(c[r] + bias));
    } else {
#pragma unroll
      for (int r = 0; r < 8; ++r) {
        if ((e0 + half * 8 + r) < n_edges)
          atomicAdd(&sums[esrc[r] * ADAPTER + ch], frelu(c[r] + bias));
      }
    }
  }
}

// ---------------------------------------------------------------------------
// Node kernel: fused down-proj -> scatter-mean gather -> fusion -> up-proj
// + residual. One 16-node tile per wave; per-wave LDS tile [16][132] (stride
// 132 => conflict-free column reads). No cross-wave LDS sharing => no
// barriers needed.
// ---------------------------------------------------------------------------
#define NODE_WAVES 4
#define LDS_STRIDE 132

__global__ __launch_bounds__(NODE_WAVES * 32) void node_adapter(
    const float* __restrict__ x,         // [N, 256]
    const float* __restrict__ W_down,    // [64, 256]
    const float* __restrict__ b_down,    // [64]
    const float* __restrict__ W_fusion,  // [64, 128]
    const float* __restrict__ b_fusion,  // [64]
    const float* __restrict__ W_up,      // [256, 64]
    const float* __restrict__ b_up,      // [256]
    const float* __restrict__ sums,      // [N, 64]
    const float* __restrict__ counts,    // [N]
    float*       __restrict__ out,       // [N, 256]
    int n_nodes)
{
  __shared__ float comb[NODE_WAVES][16][LDS_STRIDE];

  const int lane  = threadIdx.x & 31;
  const int wave  = threadIdx.x >> 5;
  const int half  = lane >> 4;
  const int idx16 = lane & 15;

  const int n0 = uni((blockIdx.x * NODE_WAVES + wave) * 16);
  if (n0 >= n_nodes) return;

  const bool full = uni(n0 + 16 <= n_nodes) != 0;  // scalar tail predicate
  float (*cw)[LDS_STRIDE] = comb[wave];

  // ---- 1) down-projection: relu(x[16,256] @ W_down^T + b_down) ------------
  v8f cd[4] = {};
  {
    const int nl = full ? (n0 + idx16) : min(n0 + idx16, n_nodes - 1);
    const float* xrow = x + nl * IN_CH + half * 2;
#pragma unroll 8
    for (int s = 0; s < 64; ++s) {          // K = 256 in steps of 4
      v2f a; a.x = xrow[s * 4 + 0]; a.y = xrow[s * 4 + 1];
      const int k = s * 4 + half * 2;
#pragma unroll
      for (int nt = 0; nt < 4; ++nt) {
        const float* wr = W_down + (nt * 16 + idx16) * IN_CH + k;
        v2f b; b.x = wr[0]; b.y = wr[1];
        cd[nt] = wmma_f32(a, b, cd[nt]);
      }
    }
  }
  // C-layout -> LDS row-major (transpose for the next GEMM's A fragments).
#pragma unroll
  for (int nt = 0; nt < 4; ++nt) {
    const int   ch   = nt * 16 + idx16;
    const float bias = b_down[ch];
#pragma unroll
    for (int r = 0; r < 8; ++r)
      cw[r + half * 8][ch] = frelu(cd[nt][r] + bias);
  }

  // ---- 2) scatter-mean gather: comb[:, 64:128] = sums/max(counts,1) -------
#pragma unroll
  for (int i = 0; i < 32; ++i) {
    const int j  = i * 32 + lane;           // 0..1023 over [16 x 64]
    const int m  = j >> 6;
    const int ch = j & 63;
    const int node = full ? (n0 + m) : min(n0 + m, n_nodes - 1);
    float cnt = counts[node];
    cnt = cnt > 1.f ? cnt : 1.f;
    cw[m][ADAPTER + ch] = sums[node * ADAPTER + ch] / cnt;
  }

  // ---- 3) fusion: relu(comb[16,128] @ W_fusion^T + b_fusion) --------------
  v8f cf[4] = {};
#pragma unroll 8
  for (int s = 0; s < 32; ++s) {            // K = 128 in steps of 4
    const int k = s * 4 + half * 2;
    v2f a; a.x = cw[idx16][k]; a.y = cw[idx16][k + 1];
#pragma unroll
    for (int nt = 0; nt < 4; ++nt) {
      const float* wr = W_fusion + (nt * 16 + idx16) * (2 * ADAPTER) + k;
      v2f b; b.x = wr[0]; b.y = wr[1];
      cf[nt] = wmma_f32(a, b, cf[nt]);
    }
  }
  // Write fused back over comb[:, 0:64] (same-wave DS ordering is enforced
  // by the compiler's DScnt waits).
#pragma unroll
  for (int nt = 0; nt < 4; ++nt) {
    const int   ch   = nt * 16 + idx16;
    const float bias = b_fusion[ch];
#pragma unroll
    for (int r = 0; r < 8; ++r)
      cw[r + half * 8][ch] = frelu(cf[nt][r] + bias);
  }

  // ---- 4) up-projection + residual: out = x + fused @ W_up^T + b_up -------
  // Each half-wave's 16 lanes write 16 consecutive channels of one node:
  // two fully-utilized 64B sectors per store instruction.
#pragma unroll 1
  for (int nt = 0; nt < 16; ++nt) {         // N = 256 in tiles of 16
    v8f cu = {};
#pragma unroll
    for (int s = 0; s < 16; ++s) {          // K = 64 in steps of 4
      const int k = s * 4 + half * 2;
      v2f a; a.x = cw[idx16][k]; a.y = cw[idx16][k + 1];
      const float* wr = W_up + (nt * 16 + idx16) * ADAPTER + k;
      v2f b; b.x = wr[0]; b.y = wr[1];
      cu = wmma_f32(a, b, cu);
    }
    const int   ch   = nt * 16 + idx16;
    const float bias = b_up[ch];
    if (full) {
#pragma unroll
      for (int r = 0; r < 8; ++r) {
        const int node = n0 + r + half * 8;
        out[node * IN_CH + ch] = x[node * IN_CH + ch] + cu[r] + bias;
      }
    } else {
#pragma unroll
      for (int r = 0; r < 8; ++r) {
        const int node = n0 + r + half * 8;
        if (node < n_nodes)
          out[node * IN_CH + ch] = x[node * IN_CH + ch] + cu[r] + bias;
      }
    }
  }
}

// ---------------------------------------------------------------------------
extern "C" void kernel_launch(void* const* d_in, const int* in_sizes, int n_in,
                              void* d_out, int out_size, void* d_ws, size_t ws_size,
                              hipStream_t stream) {
  const float* x         = (const float*)d_in[0];
  const int*   edge_idx  = (const int*)d_in[1];   // [2, E]; row 0 = src
  const float* edge_attr = (const float*)d_in[2];
  const float* W_down    = (const float*)d_in[3];
  const float* b_down    = (const float*)d_in[4];
  const float* W_time    = (const float*)d_in[5];
  const float* b_time    = (const float*)d_in[6];
  const float* W_fusion  = (const float*)d_in[7];
  const float* b_fusion  = (const float*)d_in[8];
  const float* W_up      = (const float*)d_in[9];
  const float* b_up      = (const float*)d_in[10];
  float* out = (float*)d_out;

  const int n_nodes = in_sizes[0] / IN_CH;
  const int n_edges = in_sizes[2] / EDGE_DIM;
  const int* edge_src = edge_idx;                  // first row of edge_index

  // Scratch: sums [N,64] then counts [N], zeroed every launch (graph-safe).
  float* sums   = (float*)d_ws;
  float* counts = sums + (size_t)n_nodes * ADAPTER;

  const int nz = n_nodes * (ADAPTER + 1);
  zero_kernel<<<(nz + 255) / 256, 256, 0, stream>>>(sums, nz);

  const int etiles = (n_edges + 15) / 16;          // 16 edges per wave
  edge_time_scatter<<<(etiles + 7) / 8, 256, 0, stream>>>(
      edge_attr, edge_src, W_time, b_time, sums, counts, n_edges);

  const int ntiles = (n_nodes + 15) / 16;          // 16 nodes per wave
  node_adapter<<<(ntiles + NODE_WAVES - 1) / NODE_WAVES, NODE_WAVES * 32, 0, stream>>>(
      x, W_down, b_down, W_fusion, b_fusion, W_up, b_up, sums, counts, out, n_nodes);
}